// Encoder_41686952575046
// MI455X (gfx1250) — compile-verified
//
#include <hip/hip_runtime.h>
#include <hip/hip_bf16.h>

typedef float v2f __attribute__((ext_vector_type(2)));
typedef float v8f __attribute__((ext_vector_type(8)));

// -------------------------------------------------------------------------
// Fused GEMM: H[nrows x 64] = act( X[nrows x 64] @ W[64 x 64] + bias + res )
// One wave computes a 16-row x 64-col stripe with V_WMMA_F32_16X16X4_F32.
// Block = 256 threads (8 waves) -> 128 rows per block. W staged in LDS.
// -------------------------------------------------------------------------
__global__ __launch_bounds__(256) void gcn_gemm64_kernel(
    const float* __restrict__ X, const float* __restrict__ W,
    const float* __restrict__ bias, const float* __restrict__ residual,
    float* __restrict__ H, int nrows, int do_relu)
{
    __shared__ float lW[64 * 64];
    for (int i = threadIdx.x; i < 64 * 64; i += 256) lW[i] = W[i];
    __syncthreads();

    const int wave  = threadIdx.x >> 5;           // 0..7
    const int lane  = threadIdx.x & 31;
    const int rowblk = blockIdx.x * 8 + wave;     // 16-row tile index
    const int m0 = rowblk * 16;
    if (m0 >= nrows) return;                      // whole wave exits together

    const int lrow  = lane & 15;                  // M (for A) / N (for B,D)
    const int khalf = lane >> 4;                  // 0 or 1

    v8f acc[4] = {};                              // column tiles n = 0..3
#pragma unroll 4
    for (int k = 0; k < 64; k += 4) {
        // A fragment: lane,reg r -> X[m0 + (lane&15)][k + 2*khalf + r]
        const float* arow = X + (size_t)(m0 + lrow) * 64 + k + 2 * khalf;
        v2f a; a.x = arow[0]; a.y = arow[1];
#pragma unroll
        for (int n = 0; n < 4; ++n) {
            // B fragment: lane,reg r -> W[k + 2*khalf + r][16n + (lane&15)]
            const float* bptr = &lW[(k + 2 * khalf) * 64 + n * 16 + lrow];
            v2f b; b.x = bptr[0]; b.y = bptr[64];
            acc[n] = __builtin_amdgcn_wmma_f32_16x16x4_f32(
                false, a, false, b, (short)0, acc[n], false, false);
        }
    }

    // D layout: lane,reg r -> H[m0 + r + 8*khalf][16n + (lane&15)]
#pragma unroll
    for (int n = 0; n < 4; ++n) {
        const int col = n * 16 + lrow;
        const float bv = bias ? bias[col] : 0.0f;
#pragma unroll
        for (int r = 0; r < 8; ++r) {
            const int row = m0 + r + 8 * khalf;
            float v = acc[n][r] + bv;
            if (residual) v += residual[(size_t)row * 64 + col];
            if (do_relu) v = fmaxf(v, 0.0f);
            H[(size_t)row * 64 + col] = v;
        }
    }
}

// -------------------------------------------------------------------------
// Zero a float buffer
// -------------------------------------------------------------------------
__global__ void gcn_zero_kernel(float* __restrict__ p, int n)
{
    int i = blockIdx.x * blockDim.x + threadIdx.x;
    if (i < n) p[i] = 0.0f;
}

// -------------------------------------------------------------------------
// SpMM scatter: out[dst[e]] += w[e] * h[src[e]]   (16 threads per edge,
// float4 slice each, fp32 global atomics)
// -------------------------------------------------------------------------
__global__ void gcn_spmm_kernel(const float* __restrict__ h,
                                const int* __restrict__ src,
                                const int* __restrict__ dst,
                                const float* __restrict__ ew,
                                float* __restrict__ out, int E)
{
    int t = blockIdx.x * blockDim.x + threadIdx.x;
    int e = t >> 4;
    if (e >= E) return;
    int c = (t & 15) * 4;
    float w = ew[e];
    const float4 v = *(const float4*)(h + (size_t)src[e] * 64 + c);
    float* drow = out + (size_t)dst[e] * 64 + c;
    atomicAdd(drow + 0, w * v.x);
    atomicAdd(drow + 1, w * v.y);
    atomicAdd(drow + 2, w * v.z);
    atomicAdd(drow + 3, w * v.w);
}

// -------------------------------------------------------------------------
// In-place bias (+ optional relu) over [nrows x 64]
// -------------------------------------------------------------------------
__global__ void gcn_bias_act_kernel(float* __restrict__ x,
                                    const float* __restrict__ b,
                                    int total, int do_relu)
{
    int i = blockIdx.x * blockDim.x + threadIdx.x;
    if (i >= total) return;
    float v = x[i] + b[i & 63];
    if (do_relu) v = fmaxf(v, 0.0f);
    x[i] = v;
}

// -------------------------------------------------------------------------
// seg[i] = #{ j in 1..G : start_idx[j] <= i }  (searchsorted-right semantics)
// -------------------------------------------------------------------------
__global__ void gcn_seg_kernel(const int* __restrict__ start_idx,
                               int* __restrict__ seg, int N, int G)
{
    int i = blockIdx.x * blockDim.x + threadIdx.x;
    if (i >= N) return;
    int lo = 0, hi = G;                 // search over start_idx[1..G]
    while (lo < hi) {
        int mid = (lo + hi) >> 1;
        if (start_idx[1 + mid] <= i) lo = mid + 1; else hi = mid;
    }
    seg[i] = lo;
}

// -------------------------------------------------------------------------
// Per-graph mean pooling: one block per graph, 64 threads (one per dim)
// -------------------------------------------------------------------------
__global__ void gcn_pool_kernel(const float* __restrict__ h,
                                const int* __restrict__ start_idx,
                                float* __restrict__ gmean, int G)
{
    int g = blockIdx.x;
    int d = threadIdx.x;                // 0..63
    int s = start_idx[g], e = start_idx[g + 1];
    float sum = 0.0f;
    for (int i = s; i < e; ++i) sum += h[(size_t)i * 64 + d];
    float cnt = (float)(e - s);
    gmean[(size_t)g * 64 + d] = sum / fmaxf(cnt, 1.0f);
}

// -------------------------------------------------------------------------
// Broadcast per-graph FF output back to nodes
// -------------------------------------------------------------------------
__global__ void gcn_broadcast_kernel(const float* __restrict__ outg,
                                     const int* __restrict__ seg,
                                     float* __restrict__ out, int total)
{
    int i = blockIdx.x * blockDim.x + threadIdx.x;
    if (i >= total) return;
    out[i] = outg[(size_t)seg[i >> 6] * 64 + (i & 63)];
}

// -------------------------------------------------------------------------
extern "C" void kernel_launch(void* const* d_in, const int* in_sizes, int n_in,
                              void* d_out, int out_size, void* d_ws, size_t ws_size,
                              hipStream_t stream)
{
    const float* feat  = (const float*)d_in[0];
    const float* ew    = (const float*)d_in[1];
    const float* W1    = (const float*)d_in[2];
    const float* b1    = (const float*)d_in[3];
    const float* W2    = (const float*)d_in[4];
    const float* b2    = (const float*)d_in[5];
    const float* gw1   = (const float*)d_in[6];
    const float* gb1   = (const float*)d_in[7];
    const float* gw2   = (const float*)d_in[8];
    const float* gb2   = (const float*)d_in[9];
    const float* gw3   = (const float*)d_in[10];
    const float* gb3   = (const float*)d_in[11];
    const float* gws   = (const float*)d_in[12];
    const float* gbs   = (const float*)d_in[13];
    const int*   esrc  = (const int*)d_in[14];
    const int*   edst  = (const int*)d_in[15];
    const int*   sidx  = (const int*)d_in[16];

    const int N = in_sizes[0] / 64;
    const int E = in_sizes[14];
    const int G = in_sizes[16] - 1;

    float* out = (float*)d_out;

    // workspace layout
    char* ws = (char*)d_ws;
    float* bufA  = (float*)ws;                       ws += (size_t)N * 64 * 4;
    float* bufB  = (float*)ws;                       ws += (size_t)N * 64 * 4;
    int*   seg   = (int*)ws;                         ws += (size_t)N * 4;
    float* gmean = (float*)ws;                       ws += (size_t)G * 64 * 4;
    float* z1    = (float*)ws;                       ws += (size_t)G * 64 * 4;
    float* z2    = (float*)ws;                       ws += (size_t)G * 64 * 4;
    float* z3    = (float*)ws;                       ws += (size_t)G * 64 * 4;
    float* outg  = (float*)ws;                       ws += (size_t)G * 64 * 4;

    const int ND = N * 64;
    dim3 blk(256);
    dim3 gemmN((N + 127) / 128);
    dim3 gemmG((G + 127) / 128);
    dim3 gridND((ND + 255) / 256);
    dim3 gridE((E * 16 + 255) / 256);
    dim3 gridN((N + 255) / 256);

    // ---- GCN layer 1: h1 = relu(spmm(feat @ W1) + b1) ----
    gcn_gemm64_kernel<<<gemmN, blk, 0, stream>>>(feat, W1, nullptr, nullptr, bufA, N, 0);
    gcn_zero_kernel<<<gridND, blk, 0, stream>>>(bufB, ND);
    gcn_spmm_kernel<<<gridE, blk, 0, stream>>>(bufA, esrc, edst, ew, bufB, E);
    gcn_bias_act_kernel<<<gridND, blk, 0, stream>>>(bufB, b1, ND, 1);

    // ---- GCN layer 2: h = spmm(h1 @ W2) + b2 ----
    gcn_gemm64_kernel<<<gemmN, blk, 0, stream>>>(bufB, W2, nullptr, nullptr, bufA, N, 0);
    gcn_zero_kernel<<<gridND, blk, 0, stream>>>(bufB, ND);
    gcn_spmm_kernel<<<gridE, blk, 0, stream>>>(bufA, esrc, edst, ew, bufB, E);
    gcn_bias_act_kernel<<<gridND, blk, 0, stream>>>(bufB, b2, ND, 0);

    // ---- segment ids + per-graph mean pooling ----
    gcn_seg_kernel<<<gridN, blk, 0, stream>>>(sidx, seg, N, G);
    gcn_pool_kernel<<<G, 64, 0, stream>>>(bufB, sidx, gmean, G);

    // ---- global FF on G x 64 (computed once per graph, then broadcast) ----
    gcn_gemm64_kernel<<<gemmG, blk, 0, stream>>>(gmean, gw1, gb1, nullptr, z1, G, 1);
    gcn_gemm64_kernel<<<gemmG, blk, 0, stream>>>(z1,    gw2, gb2, nullptr, z2, G, 1);
    gcn_gemm64_kernel<<<gemmG, blk, 0, stream>>>(z2,    gw3, gb3, nullptr, z3, G, 1);
    gcn_gemm64_kernel<<<gemmG, blk, 0, stream>>>(gmean, gws, gbs, z3,      outg, G, 0);

    // ---- broadcast to nodes ----
    gcn_broadcast_kernel<<<gridND, blk, 0, stream>>>(outg, seg, out, ND);
}